// RNN_simple_81235011436692
// MI455X (gfx1250) — compile-verified
//
#include <hip/hip_runtime.h>
#include <hip/hip_bf16.h>
#include <stdint.h>

// ---------------------------------------------------------------------------
// 2-layer projected LSTM (B=64, T=6000, F=80, H=128, P=32) for gfx1250.
// f32 numerics throughout via V_WMMA_F32_16X16X4_F32.
// Scan kernels: recurrent weights VGPR-resident, xi software-pipelined in
// registers (scan0), h0 staged by async global->LDS double buffering (scan1).
// ---------------------------------------------------------------------------

typedef __attribute__((ext_vector_type(2))) float v2f;
typedef __attribute__((ext_vector_type(8))) float v8f;

#define WMMA4(a, b, c) \
  __builtin_amdgcn_wmma_f32_16x16x4_f32(false, (a), false, (b), (short)0, (c), false, false)

static constexpr int BB   = 64;     // batch
static constexpr int TT   = 6000;   // time
static constexpr int F0   = 80;     // n_mels
static constexpr int HH   = 128;    // hidden
static constexpr int GG   = 512;    // 4*hidden
static constexpr int PP   = 32;     // projection
static constexpr int CH   = 600;    // time chunk
static constexpr int NCH  = TT / CH;

// workspace layout (floats)
static constexpr size_t OFF_XI0   = 0;                               // [CH][64][512]
static constexpr size_t SZ_XI0    = (size_t)CH * BB * GG;
static constexpr size_t OFF_H0SEQ = OFF_XI0 + SZ_XI0;                // [CH][64][32]
static constexpr size_t SZ_H0SEQ  = (size_t)CH * BB * PP;
static constexpr size_t OFF_H0ST  = OFF_H0SEQ + SZ_H0SEQ;            // [64][32]
static constexpr size_t OFF_C0ST  = OFF_H0ST + (size_t)BB * PP;      // [64][128]
static constexpr size_t OFF_H1ST  = OFF_C0ST + (size_t)BB * HH;      // [64][32]
static constexpr size_t OFF_C1ST  = OFF_H1ST + (size_t)BB * PP;      // [64][128]

__device__ __forceinline__ float sigm_f(float x) { return 1.0f / (1.0f + __expf(-x)); }
__device__ __forceinline__ float tanh_f(float x) { return 1.0f - 2.0f / (__expf(2.0f * x) + 1.0f); }

// ---------------------------------------------------------------------------
// Kernel 1: xi0[tc][b][g] = sum_f x[b][f][t0+tc] * Wih0[g][f] + (bih0+bhh0)[g]
// One wave per 16(batch) x 64(gate) tile. K=80 -> 20 WMMA k-steps, 4 N-tiles.
// ---------------------------------------------------------------------------
__global__ __launch_bounds__(256) void k_gemm_xi0(
    const float* __restrict__ x, const float* __restrict__ Wih0,
    const float* __restrict__ bih0, const float* __restrict__ bhh0,
    float* __restrict__ xi0, int t0, int chT) {
  const int lane   = threadIdx.x & 31;
  const int ngroup = threadIdx.x >> 5;        // 0..7 -> gate group of 64
  const int mtile  = blockIdx.x;              // 0..chT*4-1
  const int tc = mtile >> 2;
  const int b0 = (mtile & 3) << 4;
  const int t  = t0 + tc;
  const int g0 = ngroup << 6;
  const int n    = lane & 15;                 // matrix row/col within tile
  const int half = lane >> 4;

  v8f acc[4] = {};
  const float* xb = x + (size_t)(b0 + n) * F0 * TT + t;   // A row m = lane&15

#pragma unroll 4
  for (int ks = 0; ks < 20; ++ks) {
    const int ka = ks * 4 + half * 2;
    v2f a;
    a[0] = xb[(size_t)ka * TT];
    a[1] = xb[(size_t)(ka + 1) * TT];
#pragma unroll
    for (int i = 0; i < 4; ++i) {
      const int n0 = g0 + i * 16;
      const float* wp = Wih0 + (size_t)(n0 + n) * F0 + ka;
      v2f bf;
      bf[0] = wp[0];
      bf[1] = wp[1];
      acc[i] = WMMA4(a, bf, acc[i]);
    }
  }

#pragma unroll
  for (int i = 0; i < 4; ++i) {
    const int n0 = g0 + i * 16;
    const float bv = bih0[n0 + n] + bhh0[n0 + n];
    float* dst = xi0 + (size_t)(tc * BB + b0) * GG + n0 + n;
#pragma unroll
    for (int j = 0; j < 8; ++j) {
      const int m = j + half * 8;           // D layout: vgpr j, lane-half half
      dst[(size_t)m * GG] = acc[i][j] + bv;
    }
  }
}

// ---------------------------------------------------------------------------
// Kernel 2: layer-0 scan. 4 blocks (batch tiles of 16), 8 waves.
// Recurrent weights (Whh0, Whr0) resident in VGPR fragments for all T steps.
// xi0 gate inputs are software-pipelined one step ahead in registers.
// ---------------------------------------------------------------------------
__global__ __launch_bounds__(256) void k_scan0(
    const float* __restrict__ xi0, const float* __restrict__ Whh0,
    const float* __restrict__ Whr0, float* __restrict__ h0seq,
    float* __restrict__ h0st, float* __restrict__ c0st,
    int chT, int firstChunk) {
  __shared__ float lds_h[16 * 33];        // h_prev [16][32], stride 33
  __shared__ float lds_ot[16 * 132];      // o*tanh(c) [16][128], stride 132
  __shared__ float lds_part[8 * 32 * 8];  // projection partials per wave

  const int tid  = threadIdx.x;
  const int w    = tid >> 5;
  const int lane = tid & 31;
  const int n    = lane & 15;
  const int half = lane >> 4;
  const int b0   = blockIdx.x * 16;

  // Whh0 B-fragments: wave w owns columns [16w,16w+16) of each gate quadrant.
  v2f wb[4][8];
#pragma unroll
  for (int q = 0; q < 4; ++q) {
    const int n0 = q * HH + w * 16;
#pragma unroll
    for (int ks = 0; ks < 8; ++ks) {
      const int kk = ks * 4 + half * 2;
      const float* wp = Whh0 + (size_t)(n0 + n) * PP + kk;
      wb[q][ks][0] = wp[0];
      wb[q][ks][1] = wp[1];
    }
  }
  // Whr0 fragments: wave role (ntile = w&1 over P, kq = w>>1 over K=128/4)
  const int nt = w & 1, kq = w >> 1, p0 = nt * 16;
  v2f wr[8];
#pragma unroll
  for (int ks = 0; ks < 8; ++ks) {
    const int k = kq * 32 + ks * 4 + half * 2;
    const float* wp = Whr0 + (size_t)(p0 + n) * HH + k;
    wr[ks][0] = wp[0];
    wr[ks][1] = wp[1];
  }

  // cell state fragment: rows m=j+8*half, cols hcol=16w+n
  v8f cst;
  if (firstChunk) {
#pragma unroll
    for (int j = 0; j < 8; ++j) cst[j] = 0.0f;
    for (int idx = tid; idx < 16 * 32; idx += 256)
      lds_h[(idx >> 5) * 33 + (idx & 31)] = 0.0f;
  } else {
#pragma unroll
    for (int j = 0; j < 8; ++j)
      cst[j] = c0st[(size_t)(b0 + j + 8 * half) * HH + w * 16 + n];
    for (int idx = tid; idx < 16 * 32; idx += 256)
      lds_h[(idx >> 5) * 33 + (idx & 31)] = h0st[(size_t)(b0 + (idx >> 5)) * PP + (idx & 31)];
  }
  __syncthreads();

  // per-lane pointer to this lane's xi0 fragment elements for a given t
  const float* xibase = xi0 + (size_t)b0 * GG + (size_t)(8 * half) * GG + w * 16 + n;
  // software pipeline: preload t=0 gate inputs into registers
  v8f gnext[4];
#pragma unroll
  for (int q = 0; q < 4; ++q)
#pragma unroll
    for (int j = 0; j < 8; ++j)
      gnext[q][j] = xibase[(size_t)j * GG + q * HH];

  for (int t = 0; t < chT; ++t) {
    v8f gq[4];
#pragma unroll
    for (int q = 0; q < 4; ++q) gq[q] = gnext[q];

    // issue next step's gate-input loads now; consumed next iteration
    const int tn = (t + 1 < chT) ? t + 1 : t;
    const float* xin = xibase + (size_t)tn * BB * GG;
#pragma unroll
    for (int q = 0; q < 4; ++q)
#pragma unroll
      for (int j = 0; j < 8; ++j)
        gnext[q][j] = xin[(size_t)j * GG + q * HH];
    if (t + 2 < chT)  // L2 prefetch two steps ahead (global_prefetch_b8)
      __builtin_prefetch(xi0 + (size_t)(t + 2) * BB * GG + (size_t)b0 * GG + (size_t)tid * 32, 0, 1);

    // gates += h_prev @ Whh^T   (K=32 -> 8 WMMA k-steps, 4 gate tiles)
#pragma unroll
    for (int ks = 0; ks < 8; ++ks) {
      const int ka = ks * 4 + half * 2;
      v2f a;
      a[0] = lds_h[n * 33 + ka];
      a[1] = lds_h[n * 33 + ka + 1];
#pragma unroll
      for (int q = 0; q < 4; ++q) gq[q] = WMMA4(a, wb[q][ks], gq[q]);
    }

    // elementwise: i,f,g,o all live in this wave for its 16 hidden columns
    const int hcol = w * 16 + n;
#pragma unroll
    for (int j = 0; j < 8; ++j) {
      const float ig = sigm_f(gq[0][j]);
      const float fg = sigm_f(gq[1][j]);
      const float gg = tanh_f(gq[2][j]);
      const float og = sigm_f(gq[3][j]);
      const float c  = fg * cst[j] + ig * gg;
      cst[j] = c;
      lds_ot[(j + 8 * half) * 132 + hcol] = og * tanh_f(c);
    }
    __syncthreads();

    // projection partial: ot[16,128] @ Whr^T -> [16,32], K split across waves
    v8f pp = {};
#pragma unroll
    for (int ks = 0; ks < 8; ++ks) {
      const int k = kq * 32 + ks * 4 + half * 2;
      v2f a;
      a[0] = lds_ot[n * 132 + k];
      a[1] = lds_ot[n * 132 + k + 1];
      pp = WMMA4(a, wr[ks], pp);
    }
    float* ps = lds_part + ((size_t)w * 32 + lane) * 8;
#pragma unroll
    for (int j = 0; j < 8; ++j) ps[j] = pp[j];
    __syncthreads();

    if (w < 2) {  // reduce 4 K-partials, publish h to LDS + global sequence
      v8f hv;
#pragma unroll
      for (int j = 0; j < 8; ++j)
        hv[j] = lds_part[((w + 0) * 32 + lane) * 8 + j] +
                lds_part[((w + 2) * 32 + lane) * 8 + j] +
                lds_part[((w + 4) * 32 + lane) * 8 + j] +
                lds_part[((w + 6) * 32 + lane) * 8 + j];
#pragma unroll
      for (int j = 0; j < 8; ++j) {
        const int m = j + 8 * half;
        const int p = w * 16 + n;
        lds_h[m * 33 + p] = hv[j];
        h0seq[((size_t)t * BB + b0 + m) * PP + p] = hv[j];
      }
    }
    __syncthreads();
  }

  // persist states for next chunk
#pragma unroll
  for (int j = 0; j < 8; ++j)
    c0st[(size_t)(b0 + j + 8 * half) * HH + w * 16 + n] = cst[j];
  for (int idx = tid; idx < 16 * 32; idx += 256)
    h0st[(size_t)(b0 + (idx >> 5)) * PP + (idx & 31)] = lds_h[(idx >> 5) * 33 + (idx & 31)];
}

// ---------------------------------------------------------------------------
// Kernel 3: layer-1 scan + fused head. A = [h0_t | h1_prev] (K=64) against
// register-resident [Wih1|Whh1] fragments. h0_t is double-buffered in LDS via
// async global->LDS loads (ASYNCcnt path), issued one step ahead.
// ---------------------------------------------------------------------------
__global__ __launch_bounds__(256) void k_scan1(
    const float* __restrict__ h0seq, const float* __restrict__ Wih1,
    const float* __restrict__ Whh1, const float* __restrict__ bih1,
    const float* __restrict__ bhh1, const float* __restrict__ Whr1,
    const float* __restrict__ W2, const float* __restrict__ b2,
    float* __restrict__ out, float* __restrict__ h1st, float* __restrict__ c1st,
    int chT, int firstChunk, int t0) {
  __shared__ float lds_h0[2][16 * 33];    // double-buffered h0_t, stride 33
  __shared__ float lds_h1[16 * 33];       // h1_prev, stride 33
  __shared__ float lds_ot[16 * 132];
  __shared__ float lds_part[8 * 32 * 8];

  const int tid  = threadIdx.x;
  const int w    = tid >> 5;
  const int lane = tid & 31;
  const int n    = lane & 15;
  const int half = lane >> 4;
  const int b0   = blockIdx.x * 16;

  // combined [Wih1 | Whh1] B-fragments, K=64 -> 16 k-steps per gate tile
  v2f wb[4][16];
  float bias[4];
#pragma unroll
  for (int q = 0; q < 4; ++q) {
    const int n0 = q * HH + w * 16;
    bias[q] = bih1[n0 + n] + bhh1[n0 + n];
#pragma unroll
    for (int ks = 0; ks < 16; ++ks) {
      const int kk = ks * 4 + half * 2;
      const float* wp = (kk < 32) ? (Wih1 + (size_t)(n0 + n) * PP + kk)
                                  : (Whh1 + (size_t)(n0 + n) * PP + (kk - 32));
      wb[q][ks][0] = wp[0];
      wb[q][ks][1] = wp[1];
    }
  }
  const int nt = w & 1, kq = w >> 1, p0 = nt * 16;
  v2f wr[8];
#pragma unroll
  for (int ks = 0; ks < 8; ++ks) {
    const int k = kq * 32 + ks * 4 + half * 2;
    const float* wp = Whr1 + (size_t)(p0 + n) * HH + k;
    wr[ks][0] = wp[0];
    wr[ks][1] = wp[1];
  }

  v8f cst;
  if (firstChunk) {
#pragma unroll
    for (int j = 0; j < 8; ++j) cst[j] = 0.0f;
    for (int idx = tid; idx < 16 * 32; idx += 256)
      lds_h1[(idx >> 5) * 33 + (idx & 31)] = 0.0f;
  } else {
#pragma unroll
    for (int j = 0; j < 8; ++j)
      cst[j] = c1st[(size_t)(b0 + j + 8 * half) * HH + w * 16 + n];
    for (int idx = tid; idx < 16 * 32; idx += 256)
      lds_h1[(idx >> 5) * 33 + (idx & 31)] =
          h1st[(size_t)(b0 + (idx >> 5)) * PP + (idx & 31)];
  }

  // async-stage h0seq[t] into LDS buffer nb (2 instructions per wave)
  auto issue_h0 = [&](int t, int nb) {
#pragma unroll
    for (int r = 0; r < 2; ++r) {
      const int idx = tid + r * 256;                 // 0..511
      const int row = idx >> 5, col = idx & 31;
      const float* gp = h0seq + ((size_t)t * BB + b0 + row) * PP + col;
      const uint32_t loff =
          (uint32_t)(uintptr_t)&lds_h0[nb][row * 33 + col];  // LDS byte offset
      asm volatile("global_load_async_to_lds_b32 %0, %1, off"
                   :: "v"(loff), "v"(gp) : "memory");
    }
  };

  issue_h0(0, 0);  // prologue: stage t=0 into buffer 0
  const float b2v = b2[0];

  for (int t = 0; t < chT; ++t) {
    const int cur = t & 1;
    // issue next step's staging into the other buffer, then wait for the
    // 2 older async ops (current buffer) and publish across waves.
    const int tn = (t + 1 < chT) ? t + 1 : t;
    issue_h0(tn, cur ^ 1);
    asm volatile("s_wait_asynccnt 2" ::: "memory");
    __syncthreads();

    v8f gq[4];
#pragma unroll
    for (int q = 0; q < 4; ++q)
#pragma unroll
      for (int j = 0; j < 8; ++j) gq[q][j] = bias[q];

    // K = 0..31 : h0_t contribution (Wih1 fragments)
#pragma unroll
    for (int ks = 0; ks < 8; ++ks) {
      const int ka = ks * 4 + half * 2;
      v2f a;
      a[0] = lds_h0[cur][n * 33 + ka];
      a[1] = lds_h0[cur][n * 33 + ka + 1];
#pragma unroll
      for (int q = 0; q < 4; ++q) gq[q] = WMMA4(a, wb[q][ks], gq[q]);
    }
    // K = 32..63 : h1_prev contribution (Whh1 fragments)
#pragma unroll
    for (int ks = 8; ks < 16; ++ks) {
      const int ka = (ks - 8) * 4 + half * 2;
      v2f a;
      a[0] = lds_h1[n * 33 + ka];
      a[1] = lds_h1[n * 33 + ka + 1];
#pragma unroll
      for (int q = 0; q < 4; ++q) gq[q] = WMMA4(a, wb[q][ks], gq[q]);
    }

    const int hcol = w * 16 + n;
#pragma unroll
    for (int j = 0; j < 8; ++j) {
      const float ig = sigm_f(gq[0][j]);
      const float fg = sigm_f(gq[1][j]);
      const float gg = tanh_f(gq[2][j]);
      const float og = sigm_f(gq[3][j]);
      const float c  = fg * cst[j] + ig * gg;
      cst[j] = c;
      lds_ot[(j + 8 * half) * 132 + hcol] = og * tanh_f(c);
    }
    __syncthreads();

    v8f pp = {};
#pragma unroll
    for (int ks = 0; ks < 8; ++ks) {
      const int k = kq * 32 + ks * 4 + half * 2;
      v2f a;
      a[0] = lds_ot[n * 132 + k];
      a[1] = lds_ot[n * 132 + k + 1];
      pp = WMMA4(a, wr[ks], pp);
    }
    float* ps = lds_part + ((size_t)w * 32 + lane) * 8;
#pragma unroll
    for (int j = 0; j < 8; ++j) ps[j] = pp[j];
    __syncthreads();

    if (w < 2) {
#pragma unroll
      for (int j = 0; j < 8; ++j) {
        const float hv = lds_part[((w + 0) * 32 + lane) * 8 + j] +
                         lds_part[((w + 2) * 32 + lane) * 8 + j] +
                         lds_part[((w + 4) * 32 + lane) * 8 + j] +
                         lds_part[((w + 6) * 32 + lane) * 8 + j];
        lds_h1[(j + 8 * half) * 33 + w * 16 + n] = hv;  // h1_prev for t+1
      }
    }
    __syncthreads();

    // fused head: out[b, t0+t] = sigmoid(relu(relu(h1) . W2 + b2))
    if (tid < 16) {
      const float* hr = &lds_h1[tid * 33];
      float s = b2v;
#pragma unroll
      for (int p = 0; p < 32; ++p) {
        const float hp = hr[p];
        s += (hp > 0.0f ? hp : 0.0f) * W2[p];
      }
      s = (s > 0.0f) ? s : 0.0f;
      out[(size_t)(b0 + tid) * TT + t0 + t] = sigm_f(s);
    }
    __syncthreads();
  }

  // drain remaining async traffic before the block exits
  asm volatile("s_wait_asynccnt 0" ::: "memory");

#pragma unroll
  for (int j = 0; j < 8; ++j)
    c1st[(size_t)(b0 + j + 8 * half) * HH + w * 16 + n] = cst[j];
  for (int idx = tid; idx < 16 * 32; idx += 256)
    h1st[(size_t)(b0 + (idx >> 5)) * PP + (idx & 31)] =
        lds_h1[(idx >> 5) * 33 + (idx & 31)];
}

// ---------------------------------------------------------------------------
extern "C" void kernel_launch(void* const* d_in, const int* in_sizes, int n_in,
                              void* d_out, int out_size, void* d_ws, size_t ws_size,
                              hipStream_t stream) {
  const float* x    = (const float*)d_in[0];
  const float* Wih0 = (const float*)d_in[1];
  const float* Whh0 = (const float*)d_in[2];
  const float* bih0 = (const float*)d_in[3];
  const float* bhh0 = (const float*)d_in[4];
  const float* Whr0 = (const float*)d_in[5];
  const float* Wih1 = (const float*)d_in[6];
  const float* Whh1 = (const float*)d_in[7];
  const float* bih1 = (const float*)d_in[8];
  const float* bhh1 = (const float*)d_in[9];
  const float* Whr1 = (const float*)d_in[10];
  const float* W2   = (const float*)d_in[11];
  const float* b2   = (const float*)d_in[12];
  float* out = (float*)d_out;

  float* ws    = (float*)d_ws;
  float* xi0   = ws + OFF_XI0;
  float* h0seq = ws + OFF_H0SEQ;
  float* h0st  = ws + OFF_H0ST;
  float* c0st  = ws + OFF_C0ST;
  float* h1st  = ws + OFF_H1ST;
  float* c1st  = ws + OFF_C1ST;

  for (int ch = 0; ch < NCH; ++ch) {
    const int t0 = ch * CH;
    k_gemm_xi0<<<CH * 4, 256, 0, stream>>>(x, Wih0, bih0, bhh0, xi0, t0, CH);
    k_scan0<<<4, 256, 0, stream>>>(xi0, Whh0, Whr0, h0seq, h0st, c0st, CH, ch == 0);
    k_scan1<<<4, 256, 0, stream>>>(h0seq, Wih1, Whh1, bih1, bhh1, Whr1, W2, b2,
                                   out, h1st, c1st, CH, ch == 0, t0);
  }
  (void)in_sizes; (void)n_in; (void)out_size; (void)ws_size;
}